// GCN_17343077941803
// MI455X (gfx1250) — compile-verified
//
#include <hip/hip_runtime.h>

#define N_NODES   250000
#define N_EDGES   8000000
#define IN_DIM    8
#define HID_DIM   16
#define OUT_ROWS  50000          // N_NODES / 5
#define K_DIM     80             // HID_DIM * GROUP
#define OUT_FEAT  445

#define NQ_COLS   112            // columns per workgroup quarter (4 * 112 = 448 >= 445)
#define NT_PER_Q  7              // 16-wide n-tiles per quarter
#define WAVES_PB  8              // waves (M-tiles) per workgroup

typedef float v2f __attribute__((ext_vector_type(2)));
typedef float v8f __attribute__((ext_vector_type(8)));

// ---------------- degree ----------------
__global__ void k_init_deg(float* __restrict__ deg) {
    int i = blockIdx.x * blockDim.x + threadIdx.x;
    if (i < N_NODES) deg[i] = 1.0f;               // self-loop contributes 1
}

__global__ void k_count_deg(const long long* __restrict__ col, float* __restrict__ deg) {
    int e = blockIdx.x * blockDim.x + threadIdx.x;
    if (e < N_EDGES) unsafeAtomicAdd(&deg[(int)col[e]], 1.0f);
}

// ------- h = x@W1, scale by dis, init accumulator with self-loop -------
__global__ void k_h_scale(const float* __restrict__ x, const float* __restrict__ W1,
                          float* __restrict__ deg_dis,       // in: deg, out: dis (in place)
                          float* __restrict__ hs,            // [N,16] pre-scaled gather table
                          float* __restrict__ acc) {         // [N,16] accumulator (d_out hidden)
    int i = blockIdx.x * blockDim.x + threadIdx.x;
    if (i >= N_NODES) return;
    float d = __frsqrt_rn(deg_dis[i]);            // deg >= 1 always
    deg_dis[i] = d;
    float xi[IN_DIM];
#pragma unroll
    for (int k = 0; k < IN_DIM; ++k) xi[k] = x[i * IN_DIM + k];
#pragma unroll
    for (int c = 0; c < HID_DIM; ++c) {
        float s = 0.f;
#pragma unroll
        for (int k = 0; k < IN_DIM; ++k) s += xi[k] * W1[k * HID_DIM + c];  // uniform -> SMEM loads
        float v = s * d;
        hs[i * HID_DIM + c]  = v;
        acc[i * HID_DIM + c] = v;                 // self-loop term
    }
}

// ------- per-edge gather + scatter-add (L2-resident tables) -------
__global__ void k_edge_agg(const long long* __restrict__ row, const long long* __restrict__ col,
                           const float* __restrict__ hs, float* __restrict__ acc) {
    int e = blockIdx.x * blockDim.x + threadIdx.x;
    if (e >= N_EDGES) return;
    int r = (int)row[e];
    int c = (int)col[e];
    const float4* src = (const float4*)(hs + (size_t)r * HID_DIM);
    float* dst = acc + (size_t)c * HID_DIM;
#pragma unroll
    for (int q = 0; q < 4; ++q) {
        float4 v = src[q];                        // global_load_b128, hits L2
        unsafeAtomicAdd(dst + q * 4 + 0, v.x);    // global_atomic_add_f32 (no return)
        unsafeAtomicAdd(dst + q * 4 + 1, v.y);
        unsafeAtomicAdd(dst + q * 4 + 2, v.z);
        unsafeAtomicAdd(dst + q * 4 + 3, v.w);
    }
}

// ------- hidden = acc*dis + b1 (in place in d_out hidden section) -------
__global__ void k_finalize(float* __restrict__ hid, const float* __restrict__ dis,
                           const float* __restrict__ b1) {
    int idx = blockIdx.x * blockDim.x + threadIdx.x;
    if (idx >= N_NODES * HID_DIM) return;
    int i = idx >> 4, c = idx & 15;
    hid[idx] = hid[idx] * dis[i] + b1[c];
}

// ------- out = relu(hidden).reshape(50000,80) @ W2 + b2, fp32 WMMA + LDS blocking -------
// 8-wave workgroup: W2 column-quarter (80x112) staged in LDS pair-interleaved;
// each wave owns one M-tile, holds its 40 A-fragment VGPRs, sweeps 7 N-tiles.
__global__ void __launch_bounds__(256) k_gemm2(const float* __restrict__ hid,   // [50000*80] flat
                                               const float* __restrict__ W2,    // [80,445]
                                               const float* __restrict__ b2,    // [445]
                                               float* __restrict__ out) {       // [50000,445]
    __shared__ float Bl[(K_DIM / 2) * NQ_COLS * 2];   // [t=k/2][nloc][j] : 35,840 B

    const int q = blockIdx.x;                     // 0..3 column quarter

    // cooperative stage of W2 slice into LDS (all waves, before any early exit)
    for (int idx = threadIdx.x; idx < (K_DIM / 2) * NQ_COLS * 2; idx += 256) {
        int t    = idx / (NQ_COLS * 2);
        int rem  = idx - t * (NQ_COLS * 2);
        int nloc = rem >> 1;
        int j    = rem & 1;
        int gcol = q * NQ_COLS + nloc;
        gcol = gcol > (OUT_FEAT - 1) ? (OUT_FEAT - 1) : gcol;   // clamp pad columns
        Bl[idx] = W2[(size_t)(2 * t + j) * OUT_FEAT + gcol];
    }
    __syncthreads();

    const int wave  = threadIdx.x >> 5;
    const int lane  = threadIdx.x & 31;
    const int mtile = blockIdx.y * WAVES_PB + wave;
    if (mtile >= OUT_ROWS / 16) return;           // after the barrier: safe

    const int half = lane >> 4;                   // 0 or 1
    const int lid  = lane & 15;
    const int m    = mtile * 16 + lid;            // A-fragment row (M = lane&15, both halves)

    // load all A fragments once, relu fused (40 VGPRs)
    v2f a[K_DIM / 4];
#pragma unroll
    for (int s = 0; s < K_DIM / 4; ++s) {
        int kb = 4 * s + 2 * half;                // A: K = 2*(lane>>4) + vgpr
        float a0 = hid[(size_t)m * K_DIM + kb];
        float a1 = hid[(size_t)m * K_DIM + kb + 1];
        a[s].x = fmaxf(a0, 0.f);
        a[s].y = fmaxf(a1, 0.f);
    }

    for (int ntl = 0; ntl < NT_PER_Q; ++ntl) {
        v8f c = {};
#pragma unroll
        for (int s = 0; s < K_DIM / 4; ++s) {
            int t    = 2 * s + half;              // pair row: kb = 2t
            int nloc = ntl * 16 + lid;
            v2f b = *(const v2f*)&Bl[(t * NQ_COLS + nloc) * 2];   // ds_load_b64
            c = __builtin_amdgcn_wmma_f32_16x16x4_f32(
                    /*neg_a=*/false, a[s], /*neg_b=*/false, b,
                    /*c_mod=*/(short)0, c, /*reuse_a=*/false, /*reuse_b=*/false);
        }
        int n0 = q * NQ_COLS + ntl * 16 + lid;
        if (n0 < OUT_FEAT) {                      // divergence only after the WMMA chain
            float bias = b2[n0];
#pragma unroll
            for (int r = 0; r < 8; ++r) {         // C layout: M = r + 8*half, N = lane&15
                int mm = mtile * 16 + r + 8 * half;
                out[(size_t)mm * OUT_FEAT + n0] = c[r] + bias;
            }
        }
    }
}

extern "C" void kernel_launch(void* const* d_in, const int* in_sizes, int n_in,
                              void* d_out, int out_size, void* d_ws, size_t ws_size,
                              hipStream_t stream) {
    const float*     x    = (const float*)d_in[0];
    const long long* eidx = (const long long*)d_in[1];   // int64 [2, E]
    const float*     W1   = (const float*)d_in[2];
    const float*     b1   = (const float*)d_in[3];
    const float*     W2   = (const float*)d_in[4];
    const float*     b2   = (const float*)d_in[5];

    const long long* erow = eidx;              // source
    const long long* ecol = eidx + N_EDGES;    // target / aggregation index

    float* out    = (float*)d_out;                          // [50000,445] then hidden [250000,16]
    float* hidden = out + (size_t)OUT_ROWS * OUT_FEAT;      // also the scatter accumulator

    float* w   = (float*)d_ws;
    float* deg = w;                            // [N] -> becomes dis in place
    float* hs  = w + N_NODES;                  // [N,16] (1 MB offset, 16B aligned)

    k_init_deg <<<(N_NODES + 255) / 256, 256, 0, stream>>>(deg);
    k_count_deg<<<(N_EDGES + 255) / 256, 256, 0, stream>>>(ecol, deg);
    k_h_scale  <<<(N_NODES + 255) / 256, 256, 0, stream>>>(x, W1, deg, hs, hidden);
    k_edge_agg <<<(N_EDGES + 255) / 256, 256, 0, stream>>>(erow, ecol, hs, hidden);
    k_finalize <<<(N_NODES * HID_DIM + 255) / 256, 256, 0, stream>>>(hidden, deg, b1);

    dim3 gg(4, (OUT_ROWS / 16 + WAVES_PB - 1) / WAVES_PB);  // 4 x 391 workgroups
    k_gemm2<<<gg, 256, 0, stream>>>(hidden, W2, b2, out);
}